// RelationalGraphConvolutionSparse_1297080124153
// MI455X (gfx1250) — compile-verified
//
#include <hip/hip_runtime.h>

// ---------------------------------------------------------------------------
// relu( A1 @ (X @ W1) + A2 @ (X @ W2) )
//   X  : sparse [N, 256] COO (800k nnz)  -> densified into d_ws (L2-resident)
//   A* : sparse [N, N]  COO (800k nnz)   -> fp32 hardware-atomic scatter
//   W* : dense [256, 128] fp32           -> V_WMMA_F32_16X16X4_F32 GEMM with
//                                           TDM-staged A tiles in LDS
// ---------------------------------------------------------------------------

#define GCN_N       50000
#define GCN_IN      256
#define GCN_OUT     128
#define GCN_LDS_STRIDE 260   // 256 + 4 pad DWORDs -> conflict-free ds_load_b64

typedef __attribute__((ext_vector_type(2))) float v2f;
typedef __attribute__((ext_vector_type(8))) float v8f;
typedef __attribute__((ext_vector_type(4))) unsigned int u32x4;
typedef __attribute__((ext_vector_type(4))) int i32x4;
typedef __attribute__((ext_vector_type(8))) int i32x8;

#if defined(__has_builtin)
#if __has_builtin(__builtin_amdgcn_tensor_load_to_lds)
#define GCN_HAVE_TDM 1
#endif
#endif

// -------------------------------------------------- zero-fill (grid stride)
__global__ void gcn_zero_f32(float* __restrict__ p, long n) {
    long i = (long)blockIdx.x * blockDim.x + threadIdx.x;
    long stride = (long)gridDim.x * blockDim.x;
    for (; i < n; i += stride) p[i] = 0.0f;
}

// -------------------------------------------------- densify sparse X (COO)
__global__ void gcn_scatter_feat(const int* __restrict__ row,
                                 const int* __restrict__ col,
                                 const float* __restrict__ vals,
                                 float* __restrict__ Xd, int nnz) {
    int i = blockIdx.x * blockDim.x + threadIdx.x;
    if (i < nnz) {
        unsafeAtomicAdd(&Xd[(long)row[i] * GCN_IN + col[i]], vals[i]);
    }
}

// -------------------------------------------------- W pack: K-pair interleave
// Wp[(k/2)*128 + n][0,1] = W[k][n], W[k+1][n]  ->  B fragment = 1x b64 load
__global__ void gcn_pack_w(const float* __restrict__ W, float* __restrict__ Wp) {
    int t = blockIdx.x * blockDim.x + threadIdx.x;   // kpair*128 + n
    if (t < (GCN_IN / 2) * GCN_OUT) {
        int kp = t >> 7;
        int n  = t & 127;
        Wp[t * 2 + 0] = W[(kp * 2 + 0) * GCN_OUT + n];
        Wp[t * 2 + 1] = W[(kp * 2 + 1) * GCN_OUT + n];
    }
}

// -------------------------------------------------- dense GEMM via WMMA f32
// Block = 256 threads = 8 waves; block m-tile = 16 rows, wave w owns columns
// [16w, 16w+16). The 16x256 A tile is staged once per block into LDS by the
// Tensor Data Mover (pad_enable inserts 4 DWORDs/row -> 260-float stride,
// bank-conflict-free ds_load_b64 A fragments). Two accumulation chains
// (W1, W2) share A fragments and interleave to hide WMMA RAW latency.
__global__ __launch_bounds__(256) void gcn_gemm_wmma(
        const float* __restrict__ Xd,
        const float* __restrict__ Wp1, const float* __restrict__ Wp2,
        float* __restrict__ XW1, float* __restrict__ XW2) {
    __shared__ float As[16 * GCN_LDS_STRIDE];

    const int lane = threadIdx.x & 31;
    const int wave = threadIdx.x >> 5;      // 0..7 -> output column slice
    const int row0 = blockIdx.x * 16;       // m-tile base (N divisible by 16)
    const int m    = lane & 15;
    const int kh   = lane >> 4;             // which K-pair this lane holds
    const int n    = (lane & 15) + wave * 16;

#if defined(GCN_HAVE_TDM)
    if (wave == 0) {
        // --- Tensor DMA descriptor (D#): 2D tile 256x16 f32, row-padded ---
        unsigned long long ga =
            (unsigned long long)(const void*)(Xd + (long)row0 * GCN_IN);
        unsigned int lds_base = (unsigned int)(unsigned long long)(void*)As;
        u32x4 g0;
        g0[0] = 1u;                                   // count=1 (valid), user D#
        g0[1] = lds_base;                             // lds_addr (bytes)
        g0[2] = (unsigned int)(ga & 0xffffffffu);     // global_addr[31:0]
        g0[3] = (unsigned int)((ga >> 32) & 0x01ffffffu) | (2u << 30); // [56:32]+type=2
        i32x8 g1;
        g1[0] = (2 << 16)      // data_size = 4B
              | (1 << 20)      // pad_enable
              | (7 << 22)      // pad_interval: every 256 DWORDs
              | (3 << 25);     // pad_amount: 4 DWORDs  -> 260-dword row stride
        g1[1] = (int)(256u << 16);   // tensor_dim0 = 256 (bits [79:48] low half)
        g1[2] = (int)(16u << 16);    // tensor_dim1 = 16  (bits [111:80] low half)
        g1[3] = (int)(256u << 16);   // tile_dim0 = 256   (bits [127:112])
        g1[4] = 16;                  // tile_dim1 = 16    (bits [143:128])
        g1[5] = 256;                 // tensor_dim0_stride = 256 (bits [207:160])
        g1[6] = 0;
        g1[7] = 0;
        i32x4 z4 = {0, 0, 0, 0};
#if __clang_major__ >= 23
        i32x8 z8 = {0, 0, 0, 0, 0, 0, 0, 0};
        __builtin_amdgcn_tensor_load_to_lds(g0, g1, z4, z4, z8, 0);
#else
        __builtin_amdgcn_tensor_load_to_lds(g0, g1, z4, z4, 0);
#endif
        __builtin_amdgcn_s_wait_tensorcnt(0);
    }
#else
    // fallback: cooperative staging through VGPRs
    for (int t = threadIdx.x; t < 16 * GCN_IN; t += 256) {
        int r = t >> 8, c = t & 255;
        As[r * GCN_LDS_STRIDE + c] = Xd[(long)(row0 + r) * GCN_IN + c];
    }
#endif
    __syncthreads();

    __builtin_prefetch(Wp1 + n * 2, 0, 1);   // global_prefetch_b8 (L2-resident W)
    __builtin_prefetch(Wp2 + n * 2, 0, 1);

    v8f c1 = {};
    v8f c2 = {};
    for (int k0 = 0; k0 < GCN_IN; k0 += 4) {
        const int kb = k0 + kh * 2;
        v2f a;
        a.x = As[m * GCN_LDS_STRIDE + kb];          // ds_load_b64
        a.y = As[m * GCN_LDS_STRIDE + kb + 1];
        const float2 w1 = *(const float2*)(Wp1 + ((long)(kb >> 1) * GCN_OUT + n) * 2);
        const float2 w2 = *(const float2*)(Wp2 + ((long)(kb >> 1) * GCN_OUT + n) * 2);
        v2f b1; b1.x = w1.x; b1.y = w1.y;
        v2f b2; b2.x = w2.x; b2.y = w2.y;
        // (neg_a, A, neg_b, B, c_mod, C, reuse_a, reuse_b)
        c1 = __builtin_amdgcn_wmma_f32_16x16x4_f32(
                false, a, false, b1, (short)0, c1, false, false);
        c2 = __builtin_amdgcn_wmma_f32_16x16x4_f32(
                false, a, false, b2, (short)0, c2, false, false);
    }

    const int cm = kh * 8;
    const int cn = (lane & 15) + wave * 16;
    #pragma unroll
    for (int r = 0; r < 8; ++r) {
        XW1[(long)(row0 + cm + r) * GCN_OUT + cn] = c1[r];
        XW2[(long)(row0 + cm + r) * GCN_OUT + cn] = c2[r];
    }
}

// -------------------------------------------------- adjacency COO scatter
// One wave per edge; lane L handles the float4 chunk [4L, 4L+4) of the
// 128-wide row. XW rows (25.6 MB) are L2-resident -> on-chip gather.
__global__ void gcn_scatter_adj(const int* __restrict__ row,
                                const int* __restrict__ col,
                                const float* __restrict__ vals,
                                const float* __restrict__ XW,
                                float* __restrict__ acc, int nnz) {
    long t = (long)blockIdx.x * blockDim.x + threadIdx.x;
    int e    = (int)(t >> 5);
    int lane = (int)(t & 31);
    if (e < nnz) {
        float v = vals[e];
        const float4* src = (const float4*)(XW + (long)col[e] * GCN_OUT);
        float4 mc = src[lane];
        float* dst = acc + (long)row[e] * GCN_OUT + lane * 4;
        unsafeAtomicAdd(dst + 0, v * mc.x);
        unsafeAtomicAdd(dst + 1, v * mc.y);
        unsafeAtomicAdd(dst + 2, v * mc.z);
        unsafeAtomicAdd(dst + 3, v * mc.w);
    }
}

// -------------------------------------------------- final ReLU in place
__global__ void gcn_relu(float* __restrict__ p, long n) {
    long i = (long)blockIdx.x * blockDim.x + threadIdx.x;
    long stride = (long)gridDim.x * blockDim.x;
    for (; i < n; i += stride) p[i] = fmaxf(p[i], 0.0f);
}

// ---------------------------------------------------------------------------
extern "C" void kernel_launch(void* const* d_in, const int* in_sizes, int n_in,
                              void* d_out, int out_size, void* d_ws, size_t ws_size,
                              hipStream_t stream) {
    const int*   feat_row  = (const int*)  d_in[0];
    const int*   feat_col  = (const int*)  d_in[1];
    const float* feat_vals = (const float*)d_in[2];
    const int*   a1_row    = (const int*)  d_in[3];
    const int*   a1_col    = (const int*)  d_in[4];
    const float* a1_vals   = (const float*)d_in[5];
    const int*   a2_row    = (const int*)  d_in[6];
    const int*   a2_col    = (const int*)  d_in[7];
    const float* a2_vals   = (const float*)d_in[8];
    const float* W1        = (const float*)d_in[9];
    const float* W2        = (const float*)d_in[10];
    const int nnz_feat = in_sizes[0];
    const int nnz_adj  = in_sizes[3];

    // workspace layout (fp32): Xd [N,256] | XW1 [N,128] | XW2 [N,128] | Wp1 | Wp2
    float* Xd  = (float*)d_ws;
    float* XW1 = Xd  + (long)GCN_N * GCN_IN;
    float* XW2 = XW1 + (long)GCN_N * GCN_OUT;
    float* Wp1 = XW2 + (long)GCN_N * GCN_OUT;
    float* Wp2 = Wp1 + (long)GCN_IN * GCN_OUT;
    float* acc = (float*)d_out;                 // accumulate x1 + x2 directly

    const long out_elems = (long)GCN_N * GCN_OUT;

    // only the atomic-accumulated buffers need zeroing
    gcn_zero_f32<<<2048, 256, 0, stream>>>(Xd, (long)GCN_N * GCN_IN);
    gcn_zero_f32<<<1024, 256, 0, stream>>>(acc, out_elems);

    gcn_pack_w<<<(GCN_IN / 2) * GCN_OUT / 256, 256, 0, stream>>>(W1, Wp1);
    gcn_pack_w<<<(GCN_IN / 2) * GCN_OUT / 256, 256, 0, stream>>>(W2, Wp2);

    gcn_scatter_feat<<<(nnz_feat + 255) / 256, 256, 0, stream>>>(
        feat_row, feat_col, feat_vals, Xd, nnz_feat);

    gcn_gemm_wmma<<<GCN_N / 16, 256, 0, stream>>>(Xd, Wp1, Wp2, XW1, XW2);

    const long tadj = (long)nnz_adj * 32;
    const int  badj = (int)((tadj + 255) / 256);
    gcn_scatter_adj<<<badj, 256, 0, stream>>>(a1_row, a1_col, a1_vals, XW1, acc, nnz_adj);
    gcn_scatter_adj<<<badj, 256, 0, stream>>>(a2_row, a2_col, a2_vals, XW2, acc, nnz_adj);

    gcn_relu<<<1024, 256, 0, stream>>>(acc, out_elems);
}